// MultiInnerProductDecoder_44263932952906
// MI455X (gfx1250) — compile-verified
//
#include <hip/hip_runtime.h>
#include <math.h>

// ---------------------------------------------------------------------------
// MultiInnerProductDecoder on gfx1250 (MI455X)
//
// out[t,e] = sigmoid( sum_d z[src[t,e],d] * z[dst[t,e],d] * w[t,d] )
//   T=8 edge types, E=100000 edges, D=128, f32.
//
// Memory-bound gather kernel: z (51.2MB) fits in the 192MB L2, so we gather
// full 512B rows with coalesced b128 loads, stage 16-edge tiles in LDS, and
// run the reduction on the matrix pipe with V_WMMA_F32_16X16X4_F32 (full-f32
// WMMA): C(16x16) = A(16x128 = zs*w) x B(128x16 = zd^T), diag(C) = results.
// ---------------------------------------------------------------------------

typedef __attribute__((ext_vector_type(2))) float v2f;
typedef __attribute__((ext_vector_type(4))) float v4f;
typedef __attribute__((ext_vector_type(8))) float v8f;

#define DIMS 128
#define LDA 132                 // A row pad: bank = (4M + k) & 63, conflict-free
#define LDB 21                  // B row pad: stores & strided reads bank-clean
#define A_FLOATS (16 * LDA)     // 2112 floats = 8448 B
#define B_FLOATS (DIMS * LDB)   // 2688 floats = 10752 B
#define WAVE_LDS (A_FLOATS + B_FLOATS)
#define WAVES_PER_BLOCK 2       // 38.4 KB static LDS / block

__global__ __launch_bounds__(WAVES_PER_BLOCK * 32)
void mipd_wmma_kernel(const float* __restrict__ z,
                      const float* __restrict__ weight,
                      const int*  __restrict__ edge_src,
                      const int*  __restrict__ edge_dst,
                      float* __restrict__ out,
                      int T, int E, int nTiles, int tilesPerT)
{
    __shared__ float lds[WAVES_PER_BLOCK * WAVE_LDS];

    const int lane = threadIdx.x & 31;
    const int wave = threadIdx.x >> 5;
    float* __restrict__ As = &lds[wave * WAVE_LDS];
    float* __restrict__ Bs = As + A_FLOATS;

    const int waveGlobal = blockIdx.x * WAVES_PER_BLOCK + wave;
    const int waveStride = gridDim.x * WAVES_PER_BLOCK;

    // Gather assignment: lane -> (edge-in-tile, 4-float chunk parity)
    const int e_lane = lane >> 1;       // 0..15
    const int c2     = lane & 1;        // 0..1

    for (int tile = waveGlobal; tile < nTiles; tile += waveStride) {
        const int t  = tile / tilesPerT;
        const int eb = (tile - t * tilesPerT) * 16;

        // Per-lane gather row (clamped on tail tiles; extra dots are masked
        // out at the store).
        int e_idx = eb + e_lane;
        if (e_idx >= E) e_idx = E - 1;
        const int  si = edge_src[t * E + e_idx];
        const int  di = edge_dst[t * E + e_idx];
        const float* __restrict__ zs = z + (size_t)si * DIMS;
        const float* __restrict__ zd = z + (size_t)di * DIMS;
        const float* __restrict__ w  = weight + t * DIMS;

        // ---- Stage tile into LDS ------------------------------------------
        // As[e][d]   = zs[e][d] * w[d]          (row-major, LDA=132)
        // Bs[d][e]   = zd[e][d]                 (transposed,  LDB=21)
        #pragma unroll
        for (int j = 0; j < 16; ++j) {
            const int d = c2 * 4 + j * 8;
            v4f a4 = *(const v4f*)(zs + d);     // global_load_b128 (L2 hit)
            v4f w4 = *(const v4f*)(w + d);
            a4 *= w4;
            *(v4f*)(&As[e_lane * LDA + d]) = a4;  // ds_store_b128

            v4f b4 = *(const v4f*)(zd + d);     // global_load_b128 (L2 hit)
            Bs[(d + 0) * LDB + e_lane] = b4.x;  // transposed scatter stores
            Bs[(d + 1) * LDB + e_lane] = b4.y;
            Bs[(d + 2) * LDB + e_lane] = b4.z;
            Bs[(d + 3) * LDB + e_lane] = b4.w;
        }
        // LDS is wave-private; DS ops from one wave are in-order -> no barrier.

        // ---- 16x16x128 via 32 x V_WMMA_F32_16X16X4_F32 --------------------
        // A 16x4 f32 layout: lanes 0-15 row M=lane hold K,K+1; lanes 16-31
        // hold K+2,K+3.  B 4x16 mirrors it (rows striped across lanes).
        const int off = (lane >= 16) ? 2 : 0;
        const int mn  = lane & 15;

        v8f c = {};
        #pragma unroll
        for (int k = 0; k < DIMS; k += 4) {
            v2f a = *(const v2f*)(&As[mn * LDA + k + off]);   // ds_load_b64
            v2f b;
            b.x = Bs[(k + off + 0) * LDB + mn];
            b.y = Bs[(k + off + 1) * LDB + mn];
            // (neg_a, A, neg_b, B, c_mod, C, reuse_a, reuse_b)
            c = __builtin_amdgcn_wmma_f32_16x16x4_f32(
                    false, a, false, b, (short)0, c, false, false);
        }

        // ---- Extract diag(C) and apply sigmoid ----------------------------
        // C layout: VGPR r, lanes 0-15 -> (M=r,   N=lane),
        //                   lanes16-31 -> (M=r+8, N=lane-16).
        // diag(i,i): i<8  -> lane i,     c[i]
        //            i>=8 -> lane i+16,  c[i-8]
        union { v8f v; float f[8]; } u;
        u.v = c;
        const bool lo = (lane < 8);
        const bool hi = (lane >= 24);
        if (lo || hi) {
            const int r  = lo ? lane : (lane - 24);
            const int e  = lo ? lane : (lane - 16);
            const int ge = eb + e;
            if (ge < E) {
                const float x = u.f[r];
                out[t * E + ge] = 1.0f / (1.0f + __expf(-x));
            }
        }
    }
}

extern "C" void kernel_launch(void* const* d_in, const int* in_sizes, int n_in,
                              void* d_out, int out_size, void* d_ws, size_t ws_size,
                              hipStream_t stream)
{
    const float* z        = (const float*)d_in[0];   // [N_NODES, 128]
    const float* weight   = (const float*)d_in[1];   // [T, 128]
    const int*   edge_src = (const int*)d_in[2];     // [T, E]
    const int*   edge_dst = (const int*)d_in[3];     // [T, E]
    float*       out      = (float*)d_out;           // [T, E]

    int T = in_sizes[1] / DIMS;
    if (T <= 0) T = 1;
    int E = in_sizes[2] / T;
    if (E <= 0) return;

    const int tilesPerT = (E + 15) / 16;             // 6250 for E=100000
    const int nTiles    = T * tilesPerT;             // 50000
    const int blocks    = (nTiles + WAVES_PER_BLOCK - 1) / WAVES_PER_BLOCK;

    mipd_wmma_kernel<<<blocks, WAVES_PER_BLOCK * 32, 0, stream>>>(
        z, weight, edge_src, edge_dst, out, T, E, nTiles, tilesPerT);
}